// SDPASHA_80917183856887
// MI455X (gfx1250) — compile-verified
//
#include <hip/hip_runtime.h>
#include <hip/hip_bf16.h>

// SDPA single-token GQA decode for MI455X (gfx1250), flash-decoding style.
// Memory-bound: streams 256MB of f32 KV cache -> ~11us floor at 23.3 TB/s.
// QK^T uses V_WMMA_F32_16X16X4_F32 (4 GQA heads padded to M=16).

#define N_KV   8      // kv heads (groups)
#define N_REP  4      // query heads per kv head
#define NB     8      // batch
#define HD     128    // head dim
#define SEQ    4096   // max seq
#define NCHUNK 8      // seq chunks per (g,b)
#define CHUNKS 512    // seq positions per chunk
#define NWAVE  8      // waves per block (256 threads, wave32)
#define WSEQ   64     // seq positions per wave
#define NTILE  4      // 16-seq WMMA tiles per wave

typedef __attribute__((ext_vector_type(2))) float v2f;
typedef __attribute__((ext_vector_type(8))) float v8f;

__global__ __launch_bounds__(256) void attn_partial_kernel(
    const int* __restrict__ input_pos,
    const float* __restrict__ q,       // [32, 8, 1, 128]
    const float* __restrict__ knew,    // [8, 8, 1, 128]
    const float* __restrict__ vnew,    // [8, 8, 1, 128]
    const float* __restrict__ kcache,  // [8, 8, 4096, 128]
    const float* __restrict__ vcache,  // [8, 8, 4096, 128]
    const float* __restrict__ mask,    // [4096, 4096]
    float* __restrict__ O_ws,          // [32*8, NCHUNK, 128]
    float* __restrict__ M_ws,          // [32*8, NCHUNK]
    float* __restrict__ L_ws)          // [32*8, NCHUNK]
{
    __shared__ __align__(16) float qs[16][HD];            // Q rows 0..3, rows 4..15 = 0
    __shared__ __align__(16) float tile[NWAVE][16][HD];   // per-wave K/V staging (reused)
    __shared__ float sm[NWAVE][N_REP];
    __shared__ float sl[NWAVE][N_REP];

    const int tid  = threadIdx.x;
    const int wave = tid >> 5;
    const int lane = tid & 31;

    const int bid   = blockIdx.x;         // [0, 8*8*8)
    const int chunk = bid % NCHUNK;
    const int gb    = bid / NCHUNK;
    const int b     = gb % NB;
    const int g     = gb / NB;

    const int p = input_pos[0];           // single updated position (Q_LEN == 1)
    const float scale = 0.088388347648318447f;  // 1/sqrt(128)

    // ---- Stage Q for the 4 rep heads of this kv group; zero-pad rows 4..15 ----
    for (int i = tid; i < 16 * HD; i += 256) {
        int row = i >> 7;
        int d   = i & 127;
        float val = 0.0f;
        if (row < N_REP)
            val = q[((size_t)(g * N_REP + row) * NB + b) * HD + d];
        qs[row][d] = val;
    }
    __syncthreads();

    const float* kc_base = kcache + (size_t)(g * NB + b) * SEQ * HD;
    const float* vc_base = vcache + (size_t)(g * NB + b) * SEQ * HD;
    const float* kn = knew + (size_t)(g * NB + b) * HD;
    const float* vn = vnew + (size_t)(g * NB + b) * HD;
    const float* mrow = mask + (size_t)p * SEQ;

    const int s_wave = chunk * CHUNKS + wave * WSEQ;

    float m[N_REP], l[N_REP], o[N_REP][4];
    #pragma unroll
    for (int r = 0; r < N_REP; ++r) {
        m[r] = -__builtin_inff();
        l[r] = 0.0f;
        o[r][0] = o[r][1] = o[r][2] = o[r][3] = 0.0f;
    }

    for (int t = 0; t < NTILE; ++t) {
        const int s0 = s_wave + t * 16;

        // ---- Stage K tile (16 rows x 128 f32), patching row s==p with new k ----
        #pragma unroll
        for (int i = 0; i < 16; i += 2) {
            int row = i + (lane >> 4);
            int dd  = (lane & 15) * 8;
            int s   = s0 + row;
            const float* src = (s == p) ? kn : (kc_base + (size_t)s * HD);
            float4 a0 = *(const float4*)(src + dd);
            float4 a1 = *(const float4*)(src + dd + 4);
            *(float4*)&tile[wave][row][dd]     = a0;
            *(float4*)&tile[wave][row][dd + 4] = a1;
        }
        if (t + 1 < NTILE)  // speculative prefetch of next K tile
            __builtin_prefetch(kc_base + (size_t)(s0 + 16) * HD + lane * 32, 0, 0);
        asm volatile("s_wait_dscnt 0" ::: "memory");

        // ---- QK^T: 16x16x4 f32 WMMA, K-dim 128 in 32 steps ----
        // A 16x4 layout: lane L -> row M=L%16, k = {2*(L/16), 2*(L/16)+1}
        // B  4x16 layout (mirror): lane L -> col N=L%16, k = {2*(L/16), 2*(L/16)+1}
        const int rowA = lane & 15;
        const int hi2  = (lane >> 4) << 1;
        v8f c = {};
        #pragma unroll
        for (int kk = 0; kk < 32; ++kk) {
            int dim0 = kk * 4 + hi2;
            v2f a  = *(const v2f*)&qs[rowA][dim0];
            v2f bf = *(const v2f*)&tile[wave][rowA][dim0];
            c = __builtin_amdgcn_wmma_f32_16x16x4_f32(
                    false, a, false, bf, (short)0, c, false, false);
        }

        // ---- Scores: lane n (<16) holds heads r=0..3 in c[0..3] ----
        const int n = lane & 15;
        float mval = mrow[s0 + n];
        float sc[N_REP];
        #pragma unroll
        for (int r = 0; r < N_REP; ++r) {
            float sv = c[r] * scale + mval;
            sc[r] = (lane < 16) ? sv : -__builtin_inff();
        }

        // ---- Stage V tile over the same LDS region (DS ops are in-order) ----
        #pragma unroll
        for (int i = 0; i < 16; i += 2) {
            int row = i + (lane >> 4);
            int dd  = (lane & 15) * 8;
            int s   = s0 + row;
            const float* src = (s == p) ? vn : (vc_base + (size_t)s * HD);
            float4 a0 = *(const float4*)(src + dd);
            float4 a1 = *(const float4*)(src + dd + 4);
            *(float4*)&tile[wave][row][dd]     = a0;
            *(float4*)&tile[wave][row][dd + 4] = a1;
        }

        // ---- Online softmax update per head ----
        float pv[N_REP];
        #pragma unroll
        for (int r = 0; r < N_REP; ++r) {
            float tmax = sc[r];
            #pragma unroll
            for (int off = 16; off >= 1; off >>= 1)
                tmax = fmaxf(tmax, __shfl_xor(tmax, off, 32));
            float mnew = fmaxf(m[r], tmax);
            float f = __expf(m[r] - mnew);      // exp(-inf)=0 on first tile
            m[r] = mnew;
            float pr = (lane < 16) ? __expf(sc[r] - mnew) : 0.0f;
            float ps = pr;
            #pragma unroll
            for (int off = 16; off >= 1; off >>= 1)
                ps += __shfl_xor(ps, off, 32);
            l[r] = l[r] * f + ps;
            o[r][0] *= f; o[r][1] *= f; o[r][2] *= f; o[r][3] *= f;
            pv[r] = pr;
        }

        asm volatile("s_wait_dscnt 0" ::: "memory");

        // ---- P*V: each lane owns dims [lane*4, lane*4+4) for all 4 heads ----
        const int dbase = lane * 4;
        #pragma unroll
        for (int n2 = 0; n2 < 16; ++n2) {
            float4 vr = *(const float4*)&tile[wave][n2][dbase];
            #pragma unroll
            for (int r = 0; r < N_REP; ++r) {
                float pb = __shfl(pv[r], n2, 32);
                o[r][0] += pb * vr.x;
                o[r][1] += pb * vr.y;
                o[r][2] += pb * vr.z;
                o[r][3] += pb * vr.w;
            }
        }
    }

    // ---- In-block combine across the 8 waves (log-sum-exp) ----
    float* tw = &tile[wave][0][0];
    #pragma unroll
    for (int r = 0; r < N_REP; ++r) {
        float4 ov = make_float4(o[r][0], o[r][1], o[r][2], o[r][3]);
        *(float4*)(tw + r * HD + lane * 4) = ov;
        if (lane == 0) { sm[wave][r] = m[r]; sl[wave][r] = l[r]; }
    }
    __syncthreads();

    for (int item = tid; item < N_REP * HD; item += 256) {
        int r = item >> 7;
        int d = item & 127;
        float mx = -__builtin_inff();
        #pragma unroll
        for (int w = 0; w < NWAVE; ++w) mx = fmaxf(mx, sm[w][r]);
        float acc = 0.0f, lt = 0.0f;
        #pragma unroll
        for (int w = 0; w < NWAVE; ++w) {
            float wt = __expf(sm[w][r] - mx);
            acc += wt * tile[w][0][r * HD + d];  // flattened per-wave o
            lt  += wt * sl[w][r];
        }
        int hb = (g * N_REP + r) * NB + b;       // == h*NB + b
        size_t idx = (size_t)hb * NCHUNK + chunk;
        O_ws[idx * HD + d] = acc;
        if (d == 0) { M_ws[idx] = mx; L_ws[idx] = lt; }
    }
}

__global__ __launch_bounds__(128) void attn_combine_kernel(
    const float* __restrict__ O_ws,
    const float* __restrict__ M_ws,
    const float* __restrict__ L_ws,
    float* __restrict__ out)           // [8, 1, 32*128]
{
    const int hb = blockIdx.x;         // h*NB + b, [0, 256)
    const int h  = hb / NB;
    const int b  = hb % NB;
    const int d  = threadIdx.x;        // [0, 128)

    float mx = -__builtin_inff();
    #pragma unroll
    for (int i = 0; i < NCHUNK; ++i)
        mx = fmaxf(mx, M_ws[hb * NCHUNK + i]);

    float acc = 0.0f, lt = 0.0f;
    #pragma unroll
    for (int i = 0; i < NCHUNK; ++i) {
        size_t idx = (size_t)hb * NCHUNK + i;
        float wt = __expf(M_ws[idx] - mx);
        lt  += wt * L_ws[idx];
        acc += wt * O_ws[idx * HD + d];
    }
    out[(size_t)b * (32 * HD) + h * HD + d] = acc / lt;
}

extern "C" void kernel_launch(void* const* d_in, const int* in_sizes, int n_in,
                              void* d_out, int out_size, void* d_ws, size_t ws_size,
                              hipStream_t stream) {
    const int*   input_pos = (const int*)d_in[0];
    const float* q     = (const float*)d_in[1];
    const float* k     = (const float*)d_in[2];
    const float* v     = (const float*)d_in[3];
    const float* kc    = (const float*)d_in[4];
    const float* vc    = (const float*)d_in[5];
    const float* mask  = (const float*)d_in[6];
    float* out = (float*)d_out;

    // Workspace: O [256][8][128] + M [256][8] + L [256][8]  (~1.06 MB)
    float* O_ws = (float*)d_ws;
    float* M_ws = O_ws + (size_t)32 * NB * NCHUNK * HD;
    float* L_ws = M_ws + (size_t)32 * NB * NCHUNK;

    attn_partial_kernel<<<dim3(N_KV * NB * NCHUNK), 256, 0, stream>>>(
        input_pos, q, k, v, kc, vc, mask, O_ws, M_ws, L_ws);
    attn_combine_kernel<<<dim3(32 * NB), 128, 0, stream>>>(O_ws, M_ws, L_ws, out);
}